// SCHOOL_25013889532135
// MI455X (gfx1250) — compile-verified
//
#include <hip/hip_runtime.h>
#include <hip/hip_bf16.h>
#include <math.h>

// ---------------------------------------------------------------------------
// Problem constants (match reference)
// ---------------------------------------------------------------------------
#define NPTS   8192
#define FEAT   512
#define H1D    1024
#define H2D    512
#define OUTF   64
#define GH     512
#define GO     256
#define KNB    30      // K neighbors used
#define SCALEK 15

typedef float v2f __attribute__((ext_vector_type(2)));
typedef float v8f __attribute__((ext_vector_type(8)));

static __device__ __forceinline__ int imin(int a, int b) { return a < b ? a : b; }

// ---------------------------------------------------------------------------
// Generic fp32 WMMA GEMM:  C[M,N] = act( A[M,K] @ op(B) + bias )
//   TB=0: B is [K,N] row-major (ldb = row stride of B)
//   TB=1: B is [N,K] row-major, used as B^T (ldb = row stride of B)
// ACT: 0 = none, 1 = relu, 2 = tanh
// Block: 256 thr = 8 waves (wave32). Waves tiled 2(M) x 4(N); each wave owns a
// 32x64 C region as 2x4 V_WMMA_F32_16X16X4_F32 fragments (8 acc VGPRs each,
// 64 acc VGPRs total). Block tile = 64 x 256 -> the streaming A operand (adj,
// 268 MB > L2) is re-read only ceil(N/256) times.
// Fragment lane layouts per CDNA5 ISA 7.12.2:
//   A 16x4 : lane l -> M = l&15, holds K = 2*(l>>4) + {v0,v1}
//   B 4x16 : lane l -> N = l&15, holds K = 2*(l>>4) + {v0,v1}
//   C 16x16: lane l -> N = l&15, VGPR r -> M = r + 8*(l>>4)
// ---------------------------------------------------------------------------
template <bool TB, int ACT>
__global__ __launch_bounds__(256) void gemm_wmma_f32(
    const float* __restrict__ A, const float* __restrict__ B,
    const float* __restrict__ bias, float* __restrict__ C,
    int M, int N, int Kd, int lda, int ldb, int ldc)
{
    const int lane = threadIdx.x & 31;
    const int wave = threadIdx.x >> 5;
    const int wm = wave >> 2;            // 0..1
    const int wn = wave & 3;             // 0..3
    const int m0 = blockIdx.y * 64 + wm * 32;
    const int n0 = blockIdx.x * 256 + wn * 64;
    const int lh = lane >> 4;            // 0/1 -> K pair select
    const int ll = lane & 15;

    v8f acc[2][4];
    for (int tm = 0; tm < 2; ++tm)
        for (int tn = 0; tn < 4; ++tn) {
            float bv = 0.0f;
            if (bias) bv = bias[imin(n0 + tn * 16 + ll, N - 1)];
            for (int r = 0; r < 8; ++r) acc[tm][tn][r] = bv;
        }

    // Clamped addresses (out-of-range tiles compute garbage, stores guarded).
    const size_t rA0 = (size_t)imin(m0 + ll,      M - 1) * lda;
    const size_t rA1 = (size_t)imin(m0 + 16 + ll, M - 1) * lda;
    int cB[4];
    for (int tn = 0; tn < 4; ++tn) cB[tn] = imin(n0 + tn * 16 + ll, N - 1);

    for (int k = 0; k < Kd; k += 4) {
        const int ka = k + (lh << 1);
        v2f a0, a1, b[4];
        a0.x = A[rA0 + ka]; a0.y = A[rA0 + ka + 1];
        a1.x = A[rA1 + ka]; a1.y = A[rA1 + ka + 1];
        if (TB) {
            for (int tn = 0; tn < 4; ++tn) {
                b[tn].x = B[(size_t)cB[tn] * ldb + ka];
                b[tn].y = B[(size_t)cB[tn] * ldb + ka + 1];
            }
        } else {
            for (int tn = 0; tn < 4; ++tn) {
                b[tn].x = B[(size_t)ka * ldb + cB[tn]];
                b[tn].y = B[(size_t)(ka + 1) * ldb + cB[tn]];
            }
        }
        for (int tn = 0; tn < 4; ++tn) {
            acc[0][tn] = __builtin_amdgcn_wmma_f32_16x16x4_f32(false, a0, false, b[tn], (short)0, acc[0][tn], false, false);
            acc[1][tn] = __builtin_amdgcn_wmma_f32_16x16x4_f32(false, a1, false, b[tn], (short)0, acc[1][tn], false, false);
        }
    }

    for (int tm = 0; tm < 2; ++tm)
        for (int tn = 0; tn < 4; ++tn) {
            const int col = n0 + tn * 16 + ll;
            if (col >= N) continue;
            for (int r = 0; r < 8; ++r) {
                const int row = m0 + tm * 16 + lh * 8 + r;
                if (row >= M) continue;
                float v = acc[tm][tn][r];
                if (ACT == 1) v = fmaxf(v, 0.0f);
                else if (ACT == 2) v = tanhf(v);
                C[(size_t)row * ldc + col] = v;
            }
        }
}

// ---------------------------------------------------------------------------
// Row squared norms: sq[i] = sum_j X[i,j]^2
// ---------------------------------------------------------------------------
__global__ __launch_bounds__(256) void row_sqnorm(const float* __restrict__ X,
                                                  float* __restrict__ sq, int feat)
{
    __shared__ float red[256];
    const int i = blockIdx.x;
    float s = 0.0f;
    for (int j = threadIdx.x; j < feat; j += 256) {
        float v = X[(size_t)i * feat + j];
        s += v * v;
    }
    red[threadIdx.x] = s; __syncthreads();
    for (int off = 128; off > 0; off >>= 1) {
        if (threadIdx.x < (unsigned)off) red[threadIdx.x] += red[threadIdx.x + off];
        __syncthreads();
    }
    if (threadIdx.x == 0) sq[i] = red[0];
}

// ---------------------------------------------------------------------------
// Per-row exact order statistics (16th and 31st smallest distance) via radix
// bisection on float bit patterns (all distances >= 0 so uint order == float
// order). Row of 8192 distances lives in LDS (32 KB << 320 KB WGP LDS).
// ---------------------------------------------------------------------------
__global__ __launch_bounds__(256) void knn_select(const float* __restrict__ G,
                                                  const float* __restrict__ sq,
                                                  float* __restrict__ sigma,
                                                  float* __restrict__ thresh, int n)
{
    __shared__ float sd[NPTS];
    __shared__ unsigned scnt;
    const int i = blockIdx.x;
    const float sqi = sq[i];
    for (int j = threadIdx.x; j < n; j += 256) {
        float d2 = fmaxf(sqi + sq[j] - 2.0f * G[(size_t)i * n + j], 0.0f);
        sd[j] = sqrtf(d2);
    }
    __syncthreads();

    float res[2];
    const int ks[2] = {SCALEK, KNB};      // 0-indexed order statistics
    for (int t = 0; t < 2; ++t) {
        unsigned lo = 0u, hi = 0x7F800000u;
        while (lo < hi) {
            const unsigned mid = (lo + hi) >> 1;
            if (threadIdx.x == 0) scnt = 0u;
            __syncthreads();
            unsigned c = 0;
            for (int j = threadIdx.x; j < n; j += 256)
                c += (__float_as_uint(sd[j]) <= mid) ? 1u : 0u;
            atomicAdd(&scnt, c);
            __syncthreads();
            const unsigned total = scnt;
            __syncthreads();
            if (total >= (unsigned)(ks[t] + 1)) hi = mid; else lo = mid + 1u;
        }
        res[t] = __uint_as_float(lo);
    }
    if (threadIdx.x == 0) {
        sigma[i]  = fmaxf(res[0], 1e-8f);
        thresh[i] = res[1];
    }
}

// W_ij = exp(-d2_ij / (sigma_i*sigma_j)) masked by D_ij <= thresh_i, diag = 0.
// In-place over G (element-local).
__global__ __launch_bounds__(256) void affinity_kernel(float* __restrict__ W,
                                                       const float* __restrict__ sq,
                                                       const float* __restrict__ sigma,
                                                       const float* __restrict__ thresh, int n)
{
    const int j = blockIdx.x * 16 + (threadIdx.x & 15);
    const int i = blockIdx.y * 16 + (threadIdx.x >> 4);
    if (i >= n || j >= n) return;
    const size_t off = (size_t)i * n + j;
    const float g  = W[off];
    const float d2 = fmaxf(sq[i] + sq[j] - 2.0f * g, 0.0f);
    const float D  = sqrtf(d2);
    float w = expf(-d2 / (sigma[i] * sigma[j]));
    w = (D <= thresh[i]) ? w : 0.0f;
    if (i == j) w = 0.0f;
    W[off] = w;
}

// W = 0.5*(W + W^T) in place; block (by,bx) with bx>=by owns tile pair.
__global__ __launch_bounds__(256) void symmetrize_kernel(float* __restrict__ W, int n)
{
    const int bi = blockIdx.y, bj = blockIdx.x;
    if (bj < bi) return;
    const int i = bi * 16 + (threadIdx.x >> 4);
    const int j = bj * 16 + (threadIdx.x & 15);
    if (i >= n || j >= n || i > j) return;
    const size_t a = (size_t)i * n + j, b = (size_t)j * n + i;
    const float v = 0.5f * (W[a] + W[b]);
    W[a] = v; W[b] = v;
}

__global__ __launch_bounds__(256) void rowsum_dinv(const float* __restrict__ W,
                                                   float* __restrict__ dinv, int n)
{
    __shared__ float red[256];
    const int i = blockIdx.x;
    float s = 0.0f;
    for (int j = threadIdx.x; j < n; j += 256) s += W[(size_t)i * n + j];
    red[threadIdx.x] = s; __syncthreads();
    for (int off = 128; off > 0; off >>= 1) {
        if (threadIdx.x < (unsigned)off) red[threadIdx.x] += red[threadIdx.x + off];
        __syncthreads();
    }
    if (threadIdx.x == 0) dinv[i] = rsqrtf(fmaxf(red[0], 1e-8f));
}

__global__ __launch_bounds__(256) void normalize_adj(float* __restrict__ W,
                                                     const float* __restrict__ dinv, int n)
{
    const int j = blockIdx.x * 16 + (threadIdx.x & 15);
    const int i = blockIdx.y * 16 + (threadIdx.x >> 4);
    if (i >= n || j >= n) return;
    W[(size_t)i * n + j] *= dinv[i] * dinv[j];
}

// gram[r,c] = sum_k S[k,r]*S[k,c] + 1e-6*(r==c);  S is [m,64]
__global__ __launch_bounds__(256) void gram64(const float* __restrict__ S,
                                              float* __restrict__ gram, int m)
{
    __shared__ float red[256];
    const int r = blockIdx.y, c = blockIdx.x;
    float s = 0.0f;
    for (int k = threadIdx.x; k < m; k += 256)
        s += S[(size_t)k * 64 + r] * S[(size_t)k * 64 + c];
    red[threadIdx.x] = s; __syncthreads();
    for (int off = 128; off > 0; off >>= 1) {
        if (threadIdx.x < (unsigned)off) red[threadIdx.x] += red[threadIdx.x + off];
        __syncthreads();
    }
    if (threadIdx.x == 0) gram[r * 64 + c] = red[0] + ((r == c) ? 1e-6f : 0.0f);
}

// 64x64 Cholesky + lower-triangular inverse; orthw = sqrt(m)*inv(L)^T
__global__ __launch_bounds__(64) void chol_orthw(const float* __restrict__ gram,
                                                 float* __restrict__ orthw)
{
    __shared__ float L[64][64];
    __shared__ float Xi[64][64];
    const int t = threadIdx.x;
    for (int r = 0; r < 64; ++r) L[r][t] = gram[r * 64 + t];
    __syncthreads();
    for (int j = 0; j < 64; ++j) {
        if (t == j) L[j][j] = sqrtf(L[j][j]);
        __syncthreads();
        if (t > j) L[t][j] = L[t][j] / L[j][j];
        __syncthreads();
        if (t > j)
            for (int c = j + 1; c <= t; ++c) L[t][c] -= L[t][j] * L[c][j];
        __syncthreads();
    }
    // Forward substitution: column t of inv(L) (columns independent, no sync).
    for (int r = 0; r < 64; ++r) Xi[r][t] = 0.0f;
    for (int r = t; r < 64; ++r) {
        float s = (r == t) ? 1.0f : 0.0f;
        for (int k = t; k < r; ++k) s -= L[r][k] * Xi[k][t];
        Xi[r][t] = s / L[r][r];
    }
    __syncthreads();
    const float sm = sqrtf((float)NPTS);
    for (int j = 0; j < 64; ++j) orthw[t * 64 + j] = sm * Xi[j][t];   // inv(L)^T
}

// ---------------------------------------------------------------------------
// Per-row: neighbor distances -> simplex projection (100 iters) -> scatter to
// A (pre-zeroed) and sparse embs_hom = sum_j proj_j * semantic_H[nbr_j]
// ---------------------------------------------------------------------------
__global__ __launch_bounds__(64) void neighbor_proj(
    const float* __restrict__ Y, const float* __restrict__ semH,
    const long long* __restrict__ idx, const float* __restrict__ betap,
    const float* __restrict__ alphap, float* __restrict__ A,
    float* __restrict__ embs_hom, int n)
{
    __shared__ float yi[64];
    __shared__ float vals[KNB];
    __shared__ int   nb[KNB];
    const int i = blockIdx.x;
    const int t = threadIdx.x;
    yi[t] = Y[(size_t)i * 64 + t];
    if (t < KNB) nb[t] = (int)idx[(size_t)i * (KNB + 1) + 1 + t];
    __syncthreads();

    const float beta = *betap, alpha = *alphap;
    if (t < KNB) {
        const float* yn = Y + (size_t)nb[t] * 64;
        float s = 0.0f;
        for (int d = 0; d < 64; ++d) { float dd = yi[d] - yn[d]; s += dd * dd; }
        const float dfi = sqrtf(s);
        const float dxi = sqrtf(s + 1e-8f);
        vals[t] = -(dxi + beta * dfi) / (2.0f * alpha);
    }
    __syncthreads();

    if (t == 0) {   // tiny serial simplex projection (30 vals x 100 iters)
        float v0[KNB];
        float mean = 0.0f;
        for (int j = 0; j < KNB; ++j) mean += vals[j];
        mean *= (1.0f / KNB);
        float vmin = 1e30f;
        for (int j = 0; j < KNB; ++j) {
            v0[j] = vals[j] - mean + 1.0f / KNB;
            vmin = fminf(vmin, v0[j]);
        }
        float lam = 0.0f;
        for (int it = 0; it < 100; ++it) {
            float f = -1.0f, cnt = 0.0f;
            for (int j = 0; j < KNB; ++j) {
                const float v1 = v0[j] - lam;
                if (v1 > 0.0f) { f += v1; cnt += 1.0f; }
            }
            const float g = -fmaxf(cnt, 1.0f);
            lam = lam - f / g;
        }
        for (int j = 0; j < KNB; ++j)
            vals[j] = (vmin < 0.0f) ? fmaxf(v0[j] - lam, 0.0f) : v0[j];
    }
    __syncthreads();

    if (t < KNB) A[(size_t)i * n + nb[t]] = vals[t];
    float acc = 0.0f;
    for (int j = 0; j < KNB; ++j) acc += vals[j] * semH[(size_t)nb[j] * 64 + t];
    embs_hom[(size_t)i * 64 + t] = acc;
}

// ---------------------------------------------------------------------------
// Host-side GEMM dispatch
// ---------------------------------------------------------------------------
static void run_gemm(bool tb, int act, const float* A, const float* B,
                     const float* bias, float* C, int M, int N, int Kd,
                     int lda, int ldb, int ldc, hipStream_t s)
{
    dim3 grid((N + 255) / 256, (M + 63) / 64), blk(256);
    if (tb)             gemm_wmma_f32<true, 0><<<grid, blk, 0, s>>>(A, B, bias, C, M, N, Kd, lda, ldb, ldc);
    else if (act == 0)  gemm_wmma_f32<false, 0><<<grid, blk, 0, s>>>(A, B, bias, C, M, N, Kd, lda, ldb, ldc);
    else if (act == 1)  gemm_wmma_f32<false, 1><<<grid, blk, 0, s>>>(A, B, bias, C, M, N, Kd, lda, ldb, ldc);
    else                gemm_wmma_f32<false, 2><<<grid, blk, 0, s>>>(A, B, bias, C, M, N, Kd, lda, ldb, ldc);
}

extern "C" void kernel_launch(void* const* d_in, const int* in_sizes, int n_in,
                              void* d_out, int out_size, void* d_ws, size_t ws_size,
                              hipStream_t stream)
{
    const float*     x      = (const float*)d_in[0];
    const float*     x_orth = (const float*)d_in[1];
    const float*     beta   = (const float*)d_in[2];
    const float*     alpha  = (const float*)d_in[3];
    const long long* idx    = (const long long*)d_in[4];   // int64 per reference
    const float*     W1     = (const float*)d_in[5];
    const float*     b1     = (const float*)d_in[6];
    const float*     W2     = (const float*)d_in[7];
    const float*     b2     = (const float*)d_in[8];
    const float*     W3     = (const float*)d_in[9];
    const float*     b3     = (const float*)d_in[10];
    const float*     Wg1    = (const float*)d_in[11];
    const float*     Wg2    = (const float*)d_in[12];

    const int n = NPTS;

    // Output layout: embs_hom [N,64] | embs_graph [N,256] | A [N,N] | Y [N,64]
    float* out        = (float*)d_out;
    float* embs_hom   = out;
    float* embs_graph = embs_hom + (size_t)n * OUTF;
    float* Aout       = embs_graph + (size_t)n * GO;
    float* Yout       = Aout + (size_t)n * n;

    // Workspace carve-up (~331 MB)
    float* ws = (float*)d_ws;
    size_t p = 0;
    float* Wbuf  = ws + p; p += (size_t)n * n;      // G -> W -> adj (in place)
    float* buf1  = ws + p; p += (size_t)n * H1D;    // MLP hidden1 / GCN h
    float* buf2  = ws + p; p += (size_t)n * H2D;    // MLP hidden2 / GCN t1
    float* semO  = ws + p; p += (size_t)n * OUTF;   // sem_orth
    float* semH  = ws + p; p += (size_t)n * OUTF;   // semantic_H
    float* t2    = ws + p; p += (size_t)n * GO;     // h @ Wg2
    float* sq    = ws + p; p += n;
    float* sigma = ws + p; p += n;
    float* thr   = ws + p; p += n;
    float* dinv  = ws + p; p += n;
    float* gram  = ws + p; p += 64 * 64;
    float* orthw = ws + p; p += 64 * 64;

    // --- MLP on x_orth -> sem_orth ---
    run_gemm(false, 1, x_orth, W1, b1, buf1, n, H1D, FEAT, FEAT, H1D, H1D, stream);
    run_gemm(false, 1, buf1,   W2, b2, buf2, n, H2D, H1D,  H1D,  H2D, H2D, stream);
    run_gemm(false, 2, buf2,   W3, b3, semO, n, OUTF, H2D, H2D,  OUTF, OUTF, stream);

    // --- gram -> Cholesky -> orth_w ---
    gram64<<<dim3(64, 64), 256, 0, stream>>>(semO, gram, n);
    chol_orthw<<<1, 64, 0, stream>>>(gram, orthw);

    // --- MLP on x -> semantic_H ---
    run_gemm(false, 1, x,    W1, b1, buf1, n, H1D, FEAT, FEAT, H1D, H1D, stream);
    run_gemm(false, 1, buf1, W2, b2, buf2, n, H2D, H1D,  H1D,  H2D, H2D, stream);
    run_gemm(false, 2, buf2, W3, b3, semH, n, OUTF, H2D, H2D,  OUTF, OUTF, stream);

    // --- Y = semantic_H @ orth_w  (output #4) ---
    run_gemm(false, 0, semH, orthw, nullptr, Yout, n, OUTF, OUTF, OUTF, OUTF, OUTF, stream);

    // --- adjacency pipeline (single N x N buffer, in place) ---
    row_sqnorm<<<n, 256, 0, stream>>>(x, sq, FEAT);
    run_gemm(true, 0, x, x, nullptr, Wbuf, n, n, FEAT, FEAT, FEAT, n, stream);   // G = X X^T
    knn_select<<<n, 256, 0, stream>>>(Wbuf, sq, sigma, thr, n);
    {
        dim3 g(n / 16, n / 16);
        affinity_kernel<<<g, 256, 0, stream>>>(Wbuf, sq, sigma, thr, n);
        symmetrize_kernel<<<g, 256, 0, stream>>>(Wbuf, n);
        rowsum_dinv<<<n, 256, 0, stream>>>(Wbuf, dinv, n);
        normalize_adj<<<g, 256, 0, stream>>>(Wbuf, dinv, n);
    }

    // --- GCN branch ---
    run_gemm(false, 0, x,    Wg1, nullptr, buf2, n, GH, FEAT, FEAT, GH, GH, stream);   // t1
    run_gemm(false, 1, Wbuf, buf2, nullptr, buf1, n, GH, n,    n,   GH, GH, stream);   // h = relu(adj@t1)
    run_gemm(false, 0, buf1, Wg2, nullptr, t2,   n, GO, GH,   GH,   GO, GO, stream);   // t2
    run_gemm(false, 0, Wbuf, t2, nullptr, embs_graph, n, GO, n, n,  GO, GO, stream);   // adj@t2

    // --- A (zero + scatter) and sparse embs_hom ---
    hipMemsetAsync(Aout, 0, (size_t)n * n * sizeof(float), stream);
    neighbor_proj<<<n, 64, 0, stream>>>(Yout, semH, idx, beta, alpha, Aout, embs_hom, n);
}